// ScaleDotProductAttention_22093311771402
// MI455X (gfx1250) — compile-verified
//
#include <hip/hip_runtime.h>

typedef float v2f __attribute__((ext_vector_type(2)));
typedef float v8f __attribute__((ext_vector_type(8)));

#define BQ 8
#define HQ 16
#define LQ 1024
#define DQ 64

// One wave (32 lanes) per workgroup; each wave owns 16 query rows.
// LDS: 16 x 1024 f32 score/attn slab = 64 KB.
extern "C" __global__ __launch_bounds__(32)
void sdpa_wmma_f32(const float* __restrict__ q,
                   const float* __restrict__ k,
                   const float* __restrict__ v,
                   float* __restrict__ outp,
                   float* __restrict__ attnp)
{
    __shared__ float smem[16 * LQ];   // 64 KB

    const int lane = threadIdx.x & 31;
    const int l16  = lane & 15;
    const int half = lane >> 4;       // 0: K+0/1 , 1: K+2/3 (A/B frag halves)

    const int bh = blockIdx.x >> 6;          // 128 (b,h) pairs
    const int mb = blockIdx.x & 63;          // 64 row-blocks of 16
    const int m0 = mb * 16;                  // global row base for this wave

    const float* qb = q + (size_t)bh * LQ * DQ;
    const float* kb = k + (size_t)bh * LQ * DQ;   // reinterpreted [64][1024] row-major
    const float* vb = v + (size_t)bh * LQ * DQ;   // [1024][64]
    float* ob = outp  + (size_t)bh * LQ * DQ;
    float* ab = attnp + (size_t)bh * LQ * LQ;

    // ---- Preload Q A-fragments for all 16 K-steps (K=64, 4 per WMMA) ----
    // A 16x4 f32 layout: lane holds row M=l16; VGPR0/1 = K + (half*2) + {0,1}
    v2f afrag[16];
    {
        const float* qrow = qb + (size_t)(m0 + l16) * DQ;
        #pragma unroll
        for (int ks = 0; ks < 16; ++ks) {
            const int kk = ks * 4 + half * 2;
            afrag[ks] = *(const v2f*)(qrow + kk);   // kk even -> 8B aligned
        }
    }

    // ---- GEMM1: score[16x1024] = Q[16x64] * K2[64x1024], scaled + masked ----
    #pragma unroll 1
    for (int nt = 0; nt < 64; nt += 2) {            // two N-tiles -> 2 indep chains
        const int n0 = nt * 16, n1 = n0 + 16;
        v8f acc0 = {}, acc1 = {};
        #pragma unroll
        for (int ks = 0; ks < 16; ++ks) {
            const int kk = ks * 4 + half * 2;
            const float* kp = kb + (size_t)kk * LQ;
            // B 4x16 layout: lane holds col N=l16; VGPR0/1 = rows kk, kk+1
            v2f b0 = { kp[n0 + l16], kp[LQ + n0 + l16] };
            v2f b1 = { kp[n1 + l16], kp[LQ + n1 + l16] };
            acc0 = __builtin_amdgcn_wmma_f32_16x16x4_f32(false, afrag[ks], false, b0,
                                                         (short)0, acc0, false, false);
            acc1 = __builtin_amdgcn_wmma_f32_16x16x4_f32(false, afrag[ks], false, b1,
                                                         (short)0, acc1, false, false);
        }
        // C layout: element r lives at row (r + 8*half), col (n? + l16)
        #pragma unroll
        for (int r = 0; r < 8; ++r) {
            const int row_l = r + half * 8;
            const int rowg  = m0 + row_l;
            float v0 = acc0[r] * 0.125f;            // 1/sqrt(64)
            float v1 = acc1[r] * 0.125f;
            if (n0 + l16 > rowg) v0 = -1e-12f;      // faithful mask "bug"
            if (n1 + l16 > rowg) v1 = -1e-12f;
            smem[row_l * LQ + n0 + l16] = v0;
            smem[row_l * LQ + n1 + l16] = v1;
        }
    }

    // ---- Softmax per row (full 1024-wide, incl. -1e-12 entries) + attn out ----
    #pragma unroll 1
    for (int r = 0; r < 16; ++r) {
        float* row = smem + r * LQ;
        float mx = -3.0e38f;
        for (int c = lane; c < LQ; c += 32) mx = fmaxf(mx, row[c]);
        #pragma unroll
        for (int off = 16; off; off >>= 1) mx = fmaxf(mx, __shfl_xor(mx, off, 32));
        float s = 0.0f;
        for (int c = lane; c < LQ; c += 32) {
            float e = __expf(row[c] - mx);
            row[c] = e;
            s += e;
        }
        #pragma unroll
        for (int off = 16; off; off >>= 1) s += __shfl_xor(s, off, 32);
        const float inv = 1.0f / s;
        float* arow = ab + (size_t)(m0 + r) * LQ;
        for (int c = lane * 4; c < LQ; c += 128) {  // coalesced 16B per lane
            float4 t = *(float4*)(row + c);
            t.x *= inv; t.y *= inv; t.z *= inv; t.w *= inv;
            *(float4*)(row + c)  = t;               // keep normalized attn for GEMM2
            *(float4*)(arow + c) = t;               // global attn output
        }
    }

    // ---- GEMM2: out[16x64] = attn[16x1024] * V[1024x64] ----
    v8f o0 = {}, o1 = {}, o2 = {}, o3 = {};         // 4 indep chains (N tiles)
    const float* arow_lds = smem + l16 * LQ;        // A frag: row = l16
    #pragma unroll 1
    for (int kt = 0; kt < 256; ++kt) {
        const int kk = kt * 4 + half * 2;
        v2f a = *(const v2f*)(arow_lds + kk);       // 8B-aligned LDS load
        const float* vp = vb + (size_t)kk * DQ;
        v2f b0 = { vp[ 0 + l16], vp[DQ +  0 + l16] };
        v2f b1 = { vp[16 + l16], vp[DQ + 16 + l16] };
        v2f b2 = { vp[32 + l16], vp[DQ + 32 + l16] };
        v2f b3 = { vp[48 + l16], vp[DQ + 48 + l16] };
        o0 = __builtin_amdgcn_wmma_f32_16x16x4_f32(false, a, false, b0, (short)0, o0, false, false);
        o1 = __builtin_amdgcn_wmma_f32_16x16x4_f32(false, a, false, b1, (short)0, o1, false, false);
        o2 = __builtin_amdgcn_wmma_f32_16x16x4_f32(false, a, false, b2, (short)0, o2, false, false);
        o3 = __builtin_amdgcn_wmma_f32_16x16x4_f32(false, a, false, b3, (short)0, o3, false, false);
    }
    #pragma unroll
    for (int r = 0; r < 8; ++r) {
        const int row_l = r + half * 8;
        float* orow = ob + (size_t)(m0 + row_l) * DQ;
        orow[ 0 + l16] = o0[r];
        orow[16 + l16] = o1[r];
        orow[32 + l16] = o2[r];
        orow[48 + l16] = o3[r];
    }
}

extern "C" void kernel_launch(void* const* d_in, const int* in_sizes, int n_in,
                              void* d_out, int out_size, void* d_ws, size_t ws_size,
                              hipStream_t stream) {
    (void)in_sizes; (void)n_in; (void)out_size; (void)d_ws; (void)ws_size;
    const float* q = (const float*)d_in[0];
    const float* k = (const float*)d_in[1];
    const float* v = (const float*)d_in[2];
    // d_in[3] = mask (tril) — applied arithmetically (col > row) in-kernel.
    float* outp  = (float*)d_out;
    float* attnp = outp + (size_t)BQ * HQ * LQ * DQ;   // outputs concatenated: (out, attn)

    const int grid = BQ * HQ * (LQ / 16);              // 8192 single-wave workgroups
    sdpa_wmma_f32<<<grid, 32, 0, stream>>>(q, k, v, outp, attnp);
}